// BRepNetLite_15393162789168
// MI455X (gfx1250) — compile-verified
//
#include <hip/hip_runtime.h>

typedef __attribute__((ext_vector_type(2))) float v2f;
typedef __attribute__((ext_vector_type(8))) float v8f;

#define F 64
#define NCLS 10

// ---------------------------------------------------------------- utilities
__global__ void fill_zero_kernel(float* __restrict__ p, long n) {
  long i = (long)blockIdx.x * blockDim.x + threadIdx.x;
  long stride = (long)gridDim.x * blockDim.x;
  for (; i < n; i += stride) p[i] = 0.0f;
}

__global__ void degree_kernel(const long long* __restrict__ dst,
                              float* __restrict__ cnt, int ne) {
  int e = blockIdx.x * blockDim.x + threadIdx.x;
  if (e < ne) atomicAdd(&cnt[(int)dst[e]], 1.0f);
}

__global__ void invert_cnt_kernel(float* __restrict__ cnt, int n) {
  int i = blockIdx.x * blockDim.x + threadIdx.x;
  if (i < n) cnt[i] = 1.0f / fmaxf(cnt[i], 1.0f);
}

// ------------------------------------------------ neighbor scatter (mean num.)
// one wave per edge; each lane moves 2 consecutive features (256B/row coalesced)
__global__ void scatter_kernel(const float* __restrict__ h,
                               const long long* __restrict__ src,
                               const long long* __restrict__ dst,
                               float* __restrict__ agg, int ne) {
  int tid = blockIdx.x * blockDim.x + threadIdx.x;
  int e = tid >> 5;
  int lane = tid & 31;
  if (e >= ne) return;
  int s = (int)src[e];
  int d = (int)dst[e];
  v2f val = *(const v2f*)(h + (long)s * F + lane * 2);
  float* a = agg + (long)d * F + lane * 2;
  atomicAdd(a, val.x);
  atomicAdd(a + 1, val.y);
}

// ------------------------------------------- fused SAGE layer via f32 WMMA
// h_out = relu( (agg .* inv_cnt) @ Wl + h_in @ Wr + bl )
// one wave computes a 16x16 output tile, K=64 in 16 WMMA steps of K=4
__global__ void __launch_bounds__(256)
sage_wmma_kernel(const float* __restrict__ hin, const float* __restrict__ agg,
                 const float* __restrict__ inv, const float* __restrict__ Wl,
                 const float* __restrict__ bl, const float* __restrict__ Wr,
                 float* __restrict__ hout, int nodeTiles) {
  int gwave = (blockIdx.x * blockDim.x + threadIdx.x) >> 5;
  int lane = threadIdx.x & 31;
  int nodeTile = gwave >> 2;   // 4 output-feature tiles of 16 cover F=64
  int fTile = gwave & 3;
  if (nodeTile >= nodeTiles) return;

  int m = lane & 15;           // M (A) / N (B) index for this lane
  int hi = lane >> 4;          // lane half selects K pair {0,1} vs {2,3}
  int khalf = hi * 2;
  int row = nodeTile * 16 + m;
  int col = fTile * 16 + m;

  const float* aRow = agg + (long)row * F;
  const float* xRow = hin + (long)row * F;
  float ic = inv[row];

  v8f acc = {};
#pragma unroll
  for (int k0 = 0; k0 < F; k0 += 4) {
    int k = k0 + khalf;
    // A frag: mean = agg * inv_cnt  (lane holds A[m][k], A[m][k+1])
    v2f a1 = *(const v2f*)(aRow + k);
    a1.x *= ic; a1.y *= ic;
    // B frag: lane holds Wl[k][col], Wl[k+1][col]
    v2f b1 = { Wl[k * F + col], Wl[(k + 1) * F + col] };
    acc = __builtin_amdgcn_wmma_f32_16x16x4_f32(false, a1, false, b1,
                                                (short)0, acc, false, false);
    // root term: x @ Wr into the same accumulator
    v2f a2 = *(const v2f*)(xRow + k);
    v2f b2 = { Wr[k * F + col], Wr[(k + 1) * F + col] };
    acc = __builtin_amdgcn_wmma_f32_16x16x4_f32(false, a2, false, b2,
                                                (short)0, acc, false, false);
  }

  float bias = bl[col];
  int mbase = nodeTile * 16 + hi * 8;  // C/D layout: VGPR v -> M = v + hi*8
#pragma unroll
  for (int v = 0; v < 8; v++) {
    float val = acc[v] + bias;
    hout[(long)(mbase + v) * F + col] = fmaxf(val, 0.0f);
  }
}

// ------------------------------------------------ classifier + log_softmax
__global__ void classifier_kernel(const float* __restrict__ h,
                                  const float* __restrict__ Wc,
                                  const float* __restrict__ bc,
                                  float* __restrict__ out, int n) {
  int i = blockIdx.x * blockDim.x + threadIdx.x;
  if (i >= n) return;
  const float* hr = h + (long)i * F;
  float logit[NCLS];
#pragma unroll
  for (int c = 0; c < NCLS; c++) logit[c] = bc[c];
#pragma unroll
  for (int k = 0; k < F; k++) {
    float hv = hr[k];
#pragma unroll
    for (int c = 0; c < NCLS; c++) logit[c] = fmaf(hv, Wc[k * NCLS + c], logit[c]);
  }
  float mx = logit[0];
#pragma unroll
  for (int c = 1; c < NCLS; c++) mx = fmaxf(mx, logit[c]);
  float s = 0.0f;
#pragma unroll
  for (int c = 0; c < NCLS; c++) s += __expf(logit[c] - mx);
  float lse = __logf(s) + mx;
  float* o = out + (long)i * NCLS;
#pragma unroll
  for (int c = 0; c < NCLS; c++) o[c] = logit[c] - lse;
}

// ---------------------------------------------------------------- launcher
extern "C" void kernel_launch(void* const* d_in, const int* in_sizes, int n_in,
                              void* d_out, int out_size, void* d_ws, size_t ws_size,
                              hipStream_t stream) {
  (void)n_in; (void)out_size; (void)ws_size;
  const float*     x   = (const float*)d_in[0];
  const long long* ei  = (const long long*)d_in[1];
  const float* Wl1 = (const float*)d_in[2];
  const float* bl1 = (const float*)d_in[3];
  const float* Wr1 = (const float*)d_in[4];
  const float* Wl2 = (const float*)d_in[5];
  const float* bl2 = (const float*)d_in[6];
  const float* Wr2 = (const float*)d_in[7];
  const float* Wl3 = (const float*)d_in[8];
  const float* bl3 = (const float*)d_in[9];
  const float* Wr3 = (const float*)d_in[10];
  const float* Wc  = (const float*)d_in[11];
  const float* bc  = (const float*)d_in[12];
  float* outp = (float*)d_out;

  int n = in_sizes[0] / F;        // 100000
  int e = in_sizes[1] / 2;        // 1600000
  const long long* src = ei;      // edge_index[0]
  const long long* dst = ei + e;  // edge_index[1]

  float* agg = (float*)d_ws;              // N*F
  float* hA  = agg + (long)n * F;         // N*F
  float* hB  = hA + (long)n * F;          // N*F
  float* cnt = hB + (long)n * F;          // N

  // degree (computed fresh each launch; no cross-call state)
  fill_zero_kernel<<<256, 256, 0, stream>>>(cnt, (long)n);
  degree_kernel<<<(e + 255) / 256, 256, 0, stream>>>(dst, cnt, e);
  invert_cnt_kernel<<<(n + 255) / 256, 256, 0, stream>>>(cnt, n);

  int nodeTiles = (n + 15) / 16;                      // 6250 (exact)
  int gemmBlocks = (nodeTiles * 4 * 32 + 255) / 256;  // one wave per 16x16 tile
  int scatBlocks = (e * 32 + 255) / 256;              // one wave per edge

  // layer 1
  fill_zero_kernel<<<512, 256, 0, stream>>>(agg, (long)n * F);
  scatter_kernel<<<scatBlocks, 256, 0, stream>>>(x, src, dst, agg, e);
  sage_wmma_kernel<<<gemmBlocks, 256, 0, stream>>>(x, agg, cnt, Wl1, bl1, Wr1, hA, nodeTiles);
  // layer 2
  fill_zero_kernel<<<512, 256, 0, stream>>>(agg, (long)n * F);
  scatter_kernel<<<scatBlocks, 256, 0, stream>>>(hA, src, dst, agg, e);
  sage_wmma_kernel<<<gemmBlocks, 256, 0, stream>>>(hA, agg, cnt, Wl2, bl2, Wr2, hB, nodeTiles);
  // layer 3
  fill_zero_kernel<<<512, 256, 0, stream>>>(agg, (long)n * F);
  scatter_kernel<<<scatBlocks, 256, 0, stream>>>(hB, src, dst, agg, e);
  sage_wmma_kernel<<<gemmBlocks, 256, 0, stream>>>(hB, agg, cnt, Wl3, bl3, Wr3, hA, nodeTiles);
  // classifier + log_softmax
  classifier_kernel<<<(n + 255) / 256, 256, 0, stream>>>(hA, Wc, bc, outp, n);
}